// ObjectDetector_18631568130282
// MI455X (gfx1250) — compile-verified
//
#include <hip/hip_runtime.h>
#include <stdint.h>

#define K_PRE    300
#define K_OUT    10
#define NBUCKET  2048
#define CAND_CAP 640
#define N_TOT    195840   // 9*(128^2+64^2+32^2+16^2)
#define NIMG     16
#define NLVL     4

typedef __attribute__((ext_vector_type(2))) float v2f;
typedef __attribute__((ext_vector_type(8))) float v8f;

// ---------- helpers ----------

__device__ __forceinline__ int bucket_of(float s) {
    // s in (0.5, 1.0]; float bits are monotonic there.
    unsigned u = __float_as_uint(s);
    if (u >= 0x3F800000u) return NBUCKET - 1;      // s == 1.0
    unsigned k = u - 0x3F000000u;                  // 1 .. 0x7FFFFF
    return (int)(k >> 12);                         // 0 .. 2047
}

__device__ __forceinline__ void decode_box(float d0, float d1, float d2, float d3,
                                           float a0, float a1, float a2, float a3,
                                           float* bx, bool* valid) {
    float w  = a2 - a0, h = a3 - a1;
    float cx = a0 + 0.5f * w, cy = a1 + 0.5f * h;
    d0 = fminf(fmaxf(d0, -2.0f), 2.0f);
    d1 = fminf(fmaxf(d1, -2.0f), 2.0f);
    d2 = fminf(fmaxf(d2, -2.0f), 2.0f);
    d3 = fminf(fmaxf(d3, -2.0f), 2.0f);
    float px = cx + d0 * w, py = cy + d1 * h;
    float pw = w * __expf(d2), ph = h * __expf(d3);
    float x1 = fminf(fmaxf(px - 0.5f * pw, 0.0f), 1024.0f);
    float y1 = fminf(fmaxf(py - 0.5f * ph, 0.0f), 1024.0f);
    float x2 = fminf(fmaxf(px + 0.5f * pw, 0.0f), 1024.0f);
    float y2 = fminf(fmaxf(py + 0.5f * ph, 0.0f), 1024.0f);
    float bw = x2 - x1, bh = y2 - y1;
    *valid = (bw > 1.0f) && (bh > 1.0f) && (bw < 2000.0f) && (bh < 2000.0f);
    bx[0] = x1; bx[1] = y1; bx[2] = x2; bx[3] = y2;
}

// ---------- pass 1: fused sigmoid + decode + mask + LDS histogram ----------

__global__ __launch_bounds__(256)
void pass1_score(const float* __restrict__ cls,
                 const float* __restrict__ box,
                 const float4* __restrict__ anc,
                 float* __restrict__ scoresLvl,   // ws scores base + level offset
                 unsigned* __restrict__ hist,     // row = (b*4+li)*NBUCKET
                 int HW, int li) {
    __shared__ unsigned lh[NBUCKET];
    const int tid = threadIdx.x;
    for (int i = tid; i < NBUCKET; i += blockDim.x) lh[i] = 0u;
    __syncthreads();

    const int b = blockIdx.y;
    const int N = 9 * HW;
    const int chunk = (N + gridDim.x - 1) / gridDim.x;
    const int start = blockIdx.x * chunk;
    int end = start + chunk; if (end > N) end = N;

    const float* clsb = cls + (size_t)b * 9 * HW;   // (B,A,1,H,W): contiguous in n

    for (int n = start + tid; n < end; n += blockDim.x) {
        float logit = clsb[n];
        float s = 1.0f / (1.0f + __expf(-logit));
        int a  = n / HW;
        int nh = n - a * HW;
        const float* bp = box + ((size_t)(b * 9 + a) * 4) * HW + nh;
        __builtin_prefetch(clsb + n + blockDim.x, 0, 1);   // -> global_prefetch_b8
        __builtin_prefetch(bp + (size_t)4 * HW, 0, 1);
        float d0 = bp[0];
        float d1 = bp[(size_t)HW];
        float d2 = bp[2 * (size_t)HW];
        float d3 = bp[3 * (size_t)HW];
        float4 A = anc[n];
        float bx[4]; bool valid;
        decode_box(d0, d1, d2, d3, A.x, A.y, A.z, A.w, bx, &valid);
        float masked = ((s > 0.5f) && valid) ? s : -1.0f;
        scoresLvl[(size_t)b * N_TOT + n] = masked;
        if (masked > 0.0f) atomicAdd(&lh[bucket_of(masked)], 1u);
    }
    __syncthreads();
    unsigned* grow = hist + (size_t)(b * NLVL + li) * NBUCKET;
    for (int i = tid; i < NBUCKET; i += blockDim.x) {
        unsigned v = lh[i];
        if (v) atomicAdd(&grow[i], v);
    }
}

// ---------- pass 2: top-300 cutoff via exact-f32 WMMA histogram reduction ----------

__global__ __launch_bounds__(32)
void pass2_threshold(const unsigned* __restrict__ hist,
                     unsigned* __restrict__ tbArr) {
    __shared__ float gsum[16];
    const int il = blockIdx.x;
    const unsigned* H = hist + (size_t)il * NBUCKET;
    const int l  = threadIdx.x;      // one full wave32, EXEC all-ones
    const int m  = l & 15;           // A-matrix row
    const int hi = l >> 4;           // lanes 16-31 carry K=2,3

    // A(16x4) per step: row m covers buckets [128m, 128m+128) over 32 steps.
    // B = ones(4x16) => D[m][*] accumulates the exact row sum (counts < 2^24).
    v8f acc = {0.f, 0.f, 0.f, 0.f, 0.f, 0.f, 0.f, 0.f};
    v2f onev = {1.0f, 1.0f};
    for (int t = 0; t < 32; ++t) {
        int base = m * 128 + t * 4 + hi * 2;
        v2f av;
        av.x = (float)H[base];
        av.y = (float)H[base + 1];
        acc = __builtin_amdgcn_wmma_f32_16x16x4_f32(false, av, false, onev,
                                                    (short)0, acc, false, false);
    }
    // D layout: VGPR r, lanes 0-15 -> row r ; lanes 16-31 -> row r+8 (rows constant)
    if (l == 0)  for (int r = 0; r < 8; ++r) gsum[r] = acc[r];
    if (l == 16) for (int r = 0; r < 8; ++r) gsum[8 + r] = acc[r];
    __syncthreads();

    if (l == 0) {
        unsigned tb = 0, running = 0;
        bool found = false;
        for (int g = 15; g >= 0 && !found; --g) {
            unsigned gs = (unsigned)(gsum[g] + 0.5f);
            if (running + gs >= (unsigned)K_PRE) {
                for (int bkt = g * 128 + 127; bkt >= g * 128; --bkt) {
                    running += H[bkt];
                    if (running >= (unsigned)K_PRE) { tb = (unsigned)bkt; found = true; break; }
                }
            } else {
                running += gs;
            }
        }
        tbArr[il] = found ? tb : 0u;
    }
}

// ---------- pass 3: compact candidates above cutoff ----------

__global__ __launch_bounds__(256)
void pass3_collect(const float* __restrict__ scoresLvl,
                   const unsigned* __restrict__ tbArr,
                   unsigned* __restrict__ candCnt,
                   uint2* __restrict__ cand,
                   int N, int li) {
    const int b  = blockIdx.y;
    const int il = b * NLVL + li;
    const unsigned tb = tbArr[il];
    for (int n = blockIdx.x * blockDim.x + threadIdx.x; n < N;
         n += gridDim.x * blockDim.x) {
        float s = scoresLvl[(size_t)b * N_TOT + n];
        if (s > 0.0f && (unsigned)bucket_of(s) >= tb) {
            unsigned pos = atomicAdd(&candCnt[il], 1u);
            if (pos < CAND_CAP) {
                uint2 e; e.x = __float_as_uint(s); e.y = (unsigned)n;
                cand[(size_t)il * CAND_CAP + pos] = e;
            }
        }
    }
}

// ---------- pass 4: LDS bitonic top-300 + greedy NMS + output ----------

__global__ __launch_bounds__(512)
void pass4_nms(const float* __restrict__ box0, const float* __restrict__ box1,
               const float* __restrict__ box2, const float* __restrict__ box3,
               const float4* __restrict__ anc0, const float4* __restrict__ anc1,
               const float4* __restrict__ anc2, const float4* __restrict__ anc3,
               const uint2* __restrict__ cand, const unsigned* __restrict__ candCnt,
               float* __restrict__ out) {
    __shared__ unsigned long long keys[1024];
    __shared__ float4 sbox[K_PRE];
    __shared__ float  sscore[K_PRE];
    __shared__ int    skeep[K_PRE];

    const int tid = threadIdx.x;
    const int il  = blockIdx.x;
    const int b   = il >> 2;
    const int li  = il & 3;
    const int HWs[4] = {16384, 4096, 1024, 256};
    const int HW = HWs[li];
    const float*  boxp = (li == 0) ? box0 : (li == 1) ? box1 : (li == 2) ? box2 : box3;
    const float4* ancp = (li == 0) ? anc0 : (li == 1) ? anc1 : (li == 2) ? anc2 : anc3;

    unsigned nc = candCnt[il];
    if (nc > CAND_CAP) nc = CAND_CAP;

    // key = (inverted score bits, index): ascending sort == score desc, index asc
    for (int i = tid; i < 1024; i += 512) {
        unsigned long long key = 0xFFFFFFFFFFFFFFFFull;
        if (i < (int)nc) {
            uint2 e = cand[(size_t)il * CAND_CAP + i];
            key = ((unsigned long long)(0xFFFFFFFFu - e.x) << 32) | (unsigned long long)e.y;
        }
        keys[i] = key;
    }

    for (unsigned k = 2; k <= 1024; k <<= 1) {
        for (unsigned j = k >> 1; j > 0; j >>= 1) {
            __syncthreads();
            for (unsigned i = tid; i < 1024; i += 512) {
                unsigned ixj = i ^ j;
                if (ixj > i) {
                    bool up = ((i & k) == 0);
                    unsigned long long a = keys[i], c = keys[ixj];
                    if ((a > c) == up) { keys[i] = c; keys[ixj] = a; }
                }
            }
        }
    }
    __syncthreads();

    if (tid < K_PRE) {
        unsigned long long key = keys[tid];
        unsigned hi32 = (unsigned)(key >> 32);
        float sc = -1.0f;
        float4 bx = make_float4(0.f, 0.f, 0.f, 0.f);
        if (hi32 != 0xFFFFFFFFu) {
            sc = __uint_as_float(0xFFFFFFFFu - hi32);
            int n  = (int)(unsigned)(key & 0xFFFFFFFFu);
            int a  = n / HW;
            int nh = n - a * HW;
            const float* bp = boxp + ((size_t)(b * 9 + a) * 4) * HW + nh;
            float d0 = bp[0];
            float d1 = bp[(size_t)HW];
            float d2 = bp[2 * (size_t)HW];
            float d3 = bp[3 * (size_t)HW];
            float4 A = ancp[n];
            float bo[4]; bool valid;
            decode_box(d0, d1, d2, d3, A.x, A.y, A.z, A.w, bo, &valid);
            bx = make_float4(bo[0], bo[1], bo[2], bo[3]);
        }
        sbox[tid]   = bx;
        sscore[tid] = sc;
        skeep[tid]  = (sc > 0.0f) ? 1 : 0;
    }
    __syncthreads();

    // Greedy NMS: at round i, if kept, suppress all later overlapping boxes.
    for (int i = 0; i < K_PRE; ++i) {
        if (skeep[i] && tid > i && tid < K_PRE && skeep[tid]) {
            float4 bi = sbox[i], bj = sbox[tid];
            float ai = (bi.z - bi.x) * (bi.w - bi.y);
            float aj = (bj.z - bj.x) * (bj.w - bj.y);
            float x1 = fmaxf(bi.x, bj.x), y1 = fmaxf(bi.y, bj.y);
            float x2 = fminf(bi.z, bj.z), y2 = fminf(bi.w, bj.w);
            float inter = fmaxf(x2 - x1, 0.f) * fmaxf(y2 - y1, 0.f);
            float iou = inter / (ai + aj - inter + 1e-9f);
            if (iou > 0.3f) skeep[tid] = 0;
        }
        __syncthreads();
    }

    if (tid == 0) {
        const int outbase = b * (NLVL * K_OUT) + li * K_OUT;
        int cnt = 0;
        for (int t = 0; t < K_PRE && cnt < K_OUT; ++t) {
            if (skeep[t]) {
                float4 bx = sbox[t];
                float* o = out + (size_t)(outbase + cnt) * 5;
                o[0] = bx.x; o[1] = bx.y; o[2] = bx.z; o[3] = bx.w; o[4] = sscore[t];
                out[(size_t)NIMG * NLVL * K_OUT * 5 + outbase + cnt] = 1.0f;
                ++cnt;
            }
        }
        for (; cnt < K_OUT; ++cnt) {
            float* o = out + (size_t)(outbase + cnt) * 5;
            o[0] = 0.f; o[1] = 0.f; o[2] = 0.f; o[3] = 0.f; o[4] = 0.f;
            out[(size_t)NIMG * NLVL * K_OUT * 5 + outbase + cnt] = 0.0f;
        }
    }
}

// ---------- launch ----------

extern "C" void kernel_launch(void* const* d_in, const int* in_sizes, int n_in,
                              void* d_out, int out_size, void* d_ws, size_t ws_size,
                              hipStream_t stream) {
    (void)in_sizes; (void)n_in; (void)out_size; (void)ws_size;

    // setup_inputs() dict order: cls0, box0, anc0, cls1, box1, anc1, ...
    const float*  cls[4]  = {(const float*)d_in[0], (const float*)d_in[3],
                             (const float*)d_in[6], (const float*)d_in[9]};
    const float*  boxp[4] = {(const float*)d_in[1], (const float*)d_in[4],
                             (const float*)d_in[7], (const float*)d_in[10]};
    const float4* anc[4]  = {(const float4*)d_in[2], (const float4*)d_in[5],
                             (const float4*)d_in[8], (const float4*)d_in[11]};

    // workspace layout
    char* ws = (char*)d_ws;
    const size_t HIST_BYTES = (size_t)NIMG * NLVL * NBUCKET * sizeof(unsigned); // 512 KB
    unsigned* hist    = (unsigned*)ws;
    unsigned* candCnt = (unsigned*)(ws + HIST_BYTES);             // 64
    unsigned* tbArr   = candCnt + 64;                             // 64
    float*    scores  = (float*)(ws + HIST_BYTES + 1024);         // 16*195840 f32
    uint2*    cand    = (uint2*)((char*)scores + (size_t)NIMG * N_TOT * sizeof(float));

    hipMemsetAsync(ws, 0, HIST_BYTES + 512, stream);   // hist + candCnt (+tbArr)

    const int HWs[4]  = {16384, 4096, 1024, 256};
    const int LOFF[4] = {0, 147456, 184320, 193536};

    for (int li = 0; li < 4; ++li) {
        dim3 g(64, NIMG);
        pass1_score<<<g, 256, 0, stream>>>(cls[li], boxp[li], anc[li],
                                           scores + LOFF[li], hist, HWs[li], li);
    }

    pass2_threshold<<<NIMG * NLVL, 32, 0, stream>>>(hist, tbArr);

    for (int li = 0; li < 4; ++li) {
        int N = 9 * HWs[li];
        int gx = (N + 255) / 256; if (gx > 256) gx = 256;
        dim3 g(gx, NIMG);
        pass3_collect<<<g, 256, 0, stream>>>(scores + LOFF[li], tbArr,
                                             candCnt, cand, N, li);
    }

    pass4_nms<<<NIMG * NLVL, 512, 0, stream>>>(boxp[0], boxp[1], boxp[2], boxp[3],
                                               anc[0], anc[1], anc[2], anc[3],
                                               cand, candCnt, (float*)d_out);
}